// GCNLayer_3470333575494
// MI455X (gfx1250) — compile-verified
//
#include <hip/hip_runtime.h>
#include <hip/hip_bf16.h>

typedef __attribute__((ext_vector_type(2))) float v2f;
typedef __attribute__((ext_vector_type(8))) float v8f;

#define IN_FEATS  128
#define OUT_FEATS 64
#define PST 80   // paired-W LDS row stride (float2 units); 80*8 % 256 == 128
                 // -> half-waves hit disjoint 32-bank groups (conflict-free b64)

// ---------------------------------------------------------------------------
// Kernel 1: zero the accumulator (d_out) with vectorized stores.
// ---------------------------------------------------------------------------
__global__ __launch_bounds__(256) void gcn_zero(float4* __restrict__ out, int n4) {
    int i = blockIdx.x * 256 + threadIdx.x;
    if (i < n4) out[i] = make_float4(0.f, 0.f, 0.f, 0.f);
}

// ---------------------------------------------------------------------------
// Kernel 2: normh = (h @ W) * norm   via V_WMMA_F32_16X16X4_F32.
// One wave computes a 16x64 tile; 8 waves / 256 threads per block.
// W staged in LDS as K-pairs (float2) so each B-fragment is a single
// aligned ds_load_b64 straight into an even VGPR pair.
// ---------------------------------------------------------------------------
__global__ __launch_bounds__(256) void gcn_gemm_wmma(
    const float* __restrict__ h,      // [N, 128]
    const float* __restrict__ w,      // [128, 64]
    const float* __restrict__ norm,   // [N]
    float* __restrict__ normh,        // [N, 64]
    int n_nodes)
{
    __shared__ v2f WsP[(IN_FEATS / 2) * PST];   // 40 KB

    const int tid  = threadIdx.x;
    const int wave = tid >> 5;
    const int lane = tid & 31;
    const int lid  = lane & 15;
    const int half = lane >> 4;

    // Cooperative load of W into paired LDS layout:
    // WsP[kp*PST + n] = { W[2kp][n], W[2kp+1][n] }
    for (int idx = tid; idx < (IN_FEATS / 2) * OUT_FEATS; idx += 256) {
        const int kp = idx >> 6;     // 0..63
        const int n  = idx & 63;     // 0..63
        v2f val;
        val.x = w[(2 * kp + 0) * OUT_FEATS + n];
        val.y = w[(2 * kp + 1) * OUT_FEATS + n];
        WsP[kp * PST + n] = val;
    }
    __syncthreads();

    const int row0 = (blockIdx.x * 8 + wave) * 16;
    if (row0 >= n_nodes) return;   // wave-uniform; N is a multiple of 16

    // A-fragment source: lane covers row (row0+lid), cols {4k + 2*half, +1}.
    const float* arow = h + (size_t)(row0 + lid) * IN_FEATS + 2 * half;

    v8f acc0 = {}, acc1 = {}, acc2 = {}, acc3 = {};

    #pragma unroll 4
    for (int ks = 0; ks < IN_FEATS / 4; ++ks) {
        v2f a = *(const v2f*)(arow + 4 * ks);          // global_load_b64

        // B-fragment: rows K = 4ks + 2*half + {0,1}  ->  kpair = 2ks + half
        const v2f* bk = &WsP[(2 * ks + half) * PST + lid];
        v2f b0 = bk[ 0];                               // ds_load_b64 x4
        v2f b1 = bk[16];
        v2f b2 = bk[32];
        v2f b3 = bk[48];

        acc0 = __builtin_amdgcn_wmma_f32_16x16x4_f32(false, a, false, b0, (short)0, acc0, false, false);
        acc1 = __builtin_amdgcn_wmma_f32_16x16x4_f32(false, a, false, b1, (short)0, acc1, false, false);
        acc2 = __builtin_amdgcn_wmma_f32_16x16x4_f32(false, a, false, b2, (short)0, acc2, false, false);
        acc3 = __builtin_amdgcn_wmma_f32_16x16x4_f32(false, a, false, b3, (short)0, acc3, false, false);
    }

    // Epilogue: C/D layout — VGPR j holds row (row0 + j + 8*half), col = nt*16 + lid.
    #pragma unroll
    for (int j = 0; j < 8; ++j) {
        const int row = row0 + j + 8 * half;
        const float s = norm[row];
        float* orow = normh + (size_t)row * OUT_FEATS;
        orow[ 0 + lid] = acc0[j] * s;
        orow[16 + lid] = acc1[j] * s;
        orow[32 + lid] = acc2[j] * s;
        orow[48 + lid] = acc3[j] * s;
    }
}

// ---------------------------------------------------------------------------
// Kernel 3: edge scatter-add. One edge per wave32; each lane handles 2
// consecutive features (one coalesced global_load_b64 per wave, then 2
// global_atomic_add_f32). Messages + accumulator are L2-resident.
// ---------------------------------------------------------------------------
__global__ __launch_bounds__(256) void gcn_scatter(
    const float* __restrict__ normh,   // [N, 64]
    const int* __restrict__ src,
    const int* __restrict__ dst,
    float* __restrict__ agg,           // [N, 64]
    int n_edges)
{
    const int e = blockIdx.x * 8 + (threadIdx.x >> 5);
    if (e >= n_edges) return;
    const int lane = threadIdx.x & 31;

    const int s = src[e];
    const int d = dst[e];
    const float2 m = *(const float2*)(normh + (size_t)s * OUT_FEATS + 2 * lane);
    float* arow = agg + (size_t)d * OUT_FEATS + 2 * lane;

    atomicAdd(&arow[0], m.x);
    atomicAdd(&arow[1], m.y);
}

// ---------------------------------------------------------------------------
// Kernel 4: out = relu(agg + bias), in place.
// ---------------------------------------------------------------------------
__global__ __launch_bounds__(256) void gcn_finalize(
    float* __restrict__ out, const float* __restrict__ bias, int total)
{
    int i = blockIdx.x * 256 + threadIdx.x;
    if (i < total) {
        float v = out[i] + bias[i & (OUT_FEATS - 1)];
        out[i] = v > 0.f ? v : 0.f;
    }
}

// ---------------------------------------------------------------------------
extern "C" void kernel_launch(void* const* d_in, const int* in_sizes, int n_in,
                              void* d_out, int out_size, void* d_ws, size_t ws_size,
                              hipStream_t stream) {
    const float* h    = (const float*)d_in[0];   // [N,128]
    const float* w    = (const float*)d_in[1];   // [128,64]
    const float* norm = (const float*)d_in[2];   // [N,1]
    const float* bias = (const float*)d_in[3];   // [64]
    const int*   src  = (const int*)d_in[4];     // [E]
    const int*   dst  = (const int*)d_in[5];     // [E]

    float* out   = (float*)d_out;
    float* normh = (float*)d_ws;                 // N*64 floats = 25.6 MB scratch

    const int n_nodes = in_sizes[2];             // norm has N elements
    const int n_edges = in_sizes[4];
    const int total   = n_nodes * OUT_FEATS;

    // 1) zero accumulator
    gcn_zero<<<(total / 4 + 255) / 256, 256, 0, stream>>>((float4*)out, total / 4);

    // 2) normh = (h @ W) * norm   (WMMA f32)
    gcn_gemm_wmma<<<(n_nodes + 127) / 128, 256, 0, stream>>>(h, w, norm, normh, n_nodes);

    // 3) scatter-add along edges (8 edges per 256-thread block)
    gcn_scatter<<<(n_edges + 7) / 8, 256, 0, stream>>>(normh, src, dst, out, n_edges);

    // 4) bias + relu
    gcn_finalize<<<(total + 255) / 256, 256, 0, stream>>>(out, bias, total);
}